// ScalarMapping_36636071035609
// MI455X (gfx1250) — compile-verified
//
#include <hip/hip_runtime.h>

// out[b,c] = sum_d x[b,c,d] * w[c,d] + bias[c]
// B=512, C=512, HW=784. Memory-bound (0.5 FLOP/byte): stream x once, use
// f32 WMMA (16x16x4) so math matches the f32 reference exactly.
//
// Per-wave tile: 16 batches x 1 channel. 784/K4 = 196 WMMAs, unrolled x2
// with two accumulators. A layout (f32 16x4): lane=M (batch row),
// lanes 0-15 hold K=0,1 in VGPR0/1; lanes 16-31 hold K=2,3 -> one b64 load
// per lane at byte offset 8*(lane>=16). B (4x16) mirrors with lane=N, so the
// weight row loaded identically per lane replicates w across all 16 columns;
// every D column then equals the desired dot product.

typedef __attribute__((ext_vector_type(2))) float v2f;
typedef __attribute__((ext_vector_type(8))) float v8f;

#define BB  512
#define CC  512
#define HWW 784

__global__ __launch_bounds__(256) void chan_linear_wmma(
    const float* __restrict__ x, const float* __restrict__ w,
    const float* __restrict__ bias, float* __restrict__ out)
{
  const int waveInBlk = threadIdx.x >> 5;
  const int lane      = threadIdx.x & 31;
  const int t         = blockIdx.x * 8 + waveInBlk; // wave-tile id, 0..16383
  const int c         = t & (CC - 1);               // channel
  const int b0        = (t >> 9) << 4;              // batch-tile base

  const int half = lane >> 4;   // 0 -> K{0,1}, 1 -> K{2,3}
  const int row  = lane & 15;   // A-matrix row (batch within tile)

  const float* xrow = x + (size_t)(b0 + row) * (CC * (size_t)HWW)
                        + (size_t)c * HWW + 2 * half;
  const float* wrow = w + (size_t)c * HWW + 2 * half;

  v8f acc0 = {};
  v8f acc1 = {};

  // 784 = 8 * 98 : two K=4 WMMA steps per iteration.
  for (int k = 0; k < HWW; k += 8) {
    v2f a0 = *(const v2f*)(xrow + k);
    v2f b0v = *(const v2f*)(wrow + k);
    v2f a1 = *(const v2f*)(xrow + k + 4);
    v2f b1v = *(const v2f*)(wrow + k + 4);
    acc0 = __builtin_amdgcn_wmma_f32_16x16x4_f32(
        false, a0, false, b0v, (short)0, acc0, false, false);
    acc1 = __builtin_amdgcn_wmma_f32_16x16x4_f32(
        false, a1, false, b1v, (short)0, acc1, false, false);
  }

  v8f acc = acc0 + acc1;
  const float bb = bias[c];

  // D (f32 16x16): VGPR v, lanes 0-15 -> row v (col=lane); lanes 16-31 -> row v+8.
  // All columns identical; lane 0 stores rows 0-7, lane 16 stores rows 8-15.
  if (lane == 0) {
#pragma unroll
    for (int v = 0; v < 8; ++v)
      out[(size_t)(b0 + v) * CC + c] = acc[v] + bb;
  } else if (lane == 16) {
#pragma unroll
    for (int v = 0; v < 8; ++v)
      out[(size_t)(b0 + 8 + v) * CC + c] = acc[v] + bb;
  }
}

extern "C" void kernel_launch(void* const* d_in, const int* in_sizes, int n_in,
                              void* d_out, int out_size, void* d_ws, size_t ws_size,
                              hipStream_t stream) {
  const float* x    = (const float*)d_in[0]; // (512, 512, 28, 28)
  const float* w    = (const float*)d_in[1]; // (512, 784)
  const float* bias = (const float*)d_in[2]; // (512,)
  float* out        = (float*)d_out;         // (512, 512, 1) flat

  // (B/16) * C = 32 * 512 = 16384 wave-tiles; 8 waves per 256-thread block.
  dim3 grid(16384 / 8);
  dim3 block(256);
  chan_linear_wmma<<<grid, block, 0, stream>>>(x, w, bias, out);
}